// PermutationClosedStructure_53145925321280
// MI455X (gfx1250) — compile-verified
//
#include <hip/hip_runtime.h>

typedef float v2f __attribute__((ext_vector_type(2)));
typedef float v8f __attribute__((ext_vector_type(8)));

#define S_TOTAL 8192
#define D_DIM   8
#define CIN     64
#define COUT    32
#define RPERM   70

// result[s,r,o] = pooled0[s,r,:]·(W0-W1)[o,:] + total[s,:]·W1[o,:]
// One block per sample; 5 wave32s; each wave computes a 16(r) x 32(o) tile
// via chained v_wmma_f32_16x16x4_f32 with C preloaded with the base term.
__global__ __launch_bounds__(160) void pcs_wmma_kernel(
    const float* __restrict__ x,       // [S, 8, 64]
    const float* __restrict__ weight,  // [2, 32, 64]
    const int*   __restrict__ splits0, // [70, 4]
    float*       __restrict__ out)     // [S, 70, 32]
{
    __shared__ float lds_x[D_DIM * CIN];     // 512 f
    __shared__ float lds_wd[COUT * CIN];     // 2048 f  (W0 - W1)
    __shared__ float lds_total[CIN];         // 64 f
    __shared__ float lds_base[COUT];         // 32 f
    __shared__ int   lds_sp[RPERM * 4];      // 280 i

    const int s = blockIdx.x;
    const int t = threadIdx.x;
    const int nt = blockDim.x; // 160

    // ---- stage inputs into LDS ----
    const float* xs = x + s * (D_DIM * CIN);
    for (int i = t; i < D_DIM * CIN; i += nt) lds_x[i] = xs[i];
    for (int i = t; i < COUT * CIN; i += nt)
        lds_wd[i] = weight[i] - weight[COUT * CIN + i]; // W0 - W1
    for (int i = t; i < RPERM * 4; i += nt) lds_sp[i] = splits0[i];
    __syncthreads();

    // ---- total[c] = sum_i x[s,i,c] ----
    if (t < CIN) {
        float tot = 0.f;
        #pragma unroll
        for (int i = 0; i < D_DIM; ++i) tot += lds_x[i * CIN + t];
        lds_total[t] = tot;
    }
    __syncthreads();

    // ---- base[o] = total · W1[o,:] ----
    if (t < COUT) {
        const float* w1 = weight + COUT * CIN + t * CIN;
        float b = 0.f;
        #pragma unroll
        for (int c = 0; c < CIN; ++c) b += lds_total[c] * w1[c];
        lds_base[t] = b;
    }
    __syncthreads();

    // ---- per-wave 16x32 tile GEMM via WMMA f32 16x16x4 ----
    const int wave = t >> 5;        // tile index 0..4 over r
    const int lane = t & 31;
    const int half = lane >> 4;     // 0: K slice {0,1}; 1: K slice {2,3}
    const int lrow = lane & 15;     // A row / B,C,D column within tile
    const int off  = half * 2;

    const int r  = wave * 16 + lrow;
    const int rc = (r < RPERM) ? r : (RPERM - 1); // clamp tail reads
    const int i0 = lds_sp[rc * 4 + 0];
    const int i1 = lds_sp[rc * 4 + 1];
    const int i2 = lds_sp[rc * 4 + 2];
    const int i3 = lds_sp[rc * 4 + 3];

    // pooled0 row slices: p[k] holds channels {4k+off, 4k+off+1}
    v2f p[16];
    #pragma unroll
    for (int k = 0; k < 16; ++k) {
        const int cb = 4 * k + off;
        v2f a = *(const v2f*)&lds_x[i0 * CIN + cb];
        a    += *(const v2f*)&lds_x[i1 * CIN + cb];
        a    += *(const v2f*)&lds_x[i2 * CIN + cb];
        a    += *(const v2f*)&lds_x[i3 * CIN + cb];
        p[k] = a;
    }

    const int o0 = lrow;       // N-tile 0 column
    const int o1 = 16 + lrow;  // N-tile 1 column
    const float b0v = lds_base[o0];
    const float b1v = lds_base[o1];
    v8f acc0 = { b0v, b0v, b0v, b0v, b0v, b0v, b0v, b0v };
    v8f acc1 = { b1v, b1v, b1v, b1v, b1v, b1v, b1v, b1v };

    #pragma unroll
    for (int k = 0; k < 16; ++k) {
        const int cb = 4 * k + off;
        const v2f b0 = *(const v2f*)&lds_wd[o0 * CIN + cb];
        const v2f b1 = *(const v2f*)&lds_wd[o1 * CIN + cb];
        acc0 = __builtin_amdgcn_wmma_f32_16x16x4_f32(
            false, p[k], false, b0, (short)0, acc0, false, false);
        acc1 = __builtin_amdgcn_wmma_f32_16x16x4_f32(
            false, p[k], false, b1, (short)0, acc1, false, false);
    }

    // ---- store D: VGPR j -> row M = j + half*8, col N = lrow ----
    float* outs = out + s * (RPERM * COUT);
    #pragma unroll
    for (int j = 0; j < 8; ++j) {
        const int rl = wave * 16 + j + half * 8;
        if (rl < RPERM) {
            outs[rl * COUT + o0] = acc0[j];
            outs[rl * COUT + o1] = acc1[j];
        }
    }
}

extern "C" void kernel_launch(void* const* d_in, const int* in_sizes, int n_in,
                              void* d_out, int out_size, void* d_ws, size_t ws_size,
                              hipStream_t stream) {
    const float* x       = (const float*)d_in[0];
    const float* weight  = (const float*)d_in[1];
    const int*   splits0 = (const int*)d_in[2];
    // splits1 (d_in[3]) is unneeded: splits1 rows are complements of splits0 rows.
    float* out = (float*)d_out;

    const int samples = in_sizes[0] / (D_DIM * CIN); // 8192
    pcs_wmma_kernel<<<samples, 160, 0, stream>>>(x, weight, splits0, out);
}